// SNN_4810363373003
// MI455X (gfx1250) — compile-verified
//
#include <hip/hip_runtime.h>

typedef __attribute__((ext_vector_type(2))) float v2f;
typedef __attribute__((ext_vector_type(8))) float v8f;

#define THRESH 1.0f
#define KC     32      // K-panel depth staged in LDS
#define PROW   36      // padded LDS row stride (floats): 144B = 16B-aligned, bank-staggered

__global__ __launch_bounds__(256) void snn_zero_state(float* __restrict__ p, long n) {
    long i = (long)blockIdx.x * blockDim.x + threadIdx.x;
    if (i < n) p[i] = 0.0f;
}

// Async-copy one 64-row x KC-col fp32 weight panel (rows n0..n0+63, cols k0..k0+KC-1)
// into LDS at byte offset lbase. 256 threads x 2 transfers x 16B = 8KB.
__device__ __forceinline__ void panel_load_async(const float* __restrict__ W, int K,
                                                 int n0, int k0, unsigned lbase, int tid) {
#pragma unroll
    for (int i = 0; i < 2; ++i) {
        const int idx = tid + (i << 8);
        const int row = idx >> 3;
        const int cg  = (idx & 7) << 2;                 // column group of 4 floats (16B)
        const float* g  = W + (size_t)(n0 + row) * K + k0 + cg;
        const unsigned l = lbase + (unsigned)((row * PROW + cg) << 2);
        asm volatile("global_load_async_to_lds_b128 %0, %1, off"
                     :: "v"(l), "v"(g) : "memory");
    }
}

// One fused step:  CUR = A(MxK) @ W(NxK)^T + bias;  LIF update of mem(MxN) -> spikes.
// Block = 8 wave32s sharing one 64-col N-strip; each wave owns a 16x64 output strip
// built from 4x V_WMMA_F32_16X16X4_F32 accumulators. Weights double-buffered in LDS
// via global_load_async_to_lds_b128 (ASYNCcnt), fragments read with ds_load_b64.
__global__ __launch_bounds__(256) void snn_lif_gemm_step(
    const float* __restrict__ A,        // (M x K) row-major activations
    const float* __restrict__ W,        // (N x K) row-major weights (B-matrix = W^T)
    const float* __restrict__ bias,     // (N)
    const float* __restrict__ beta_p,   // device scalar
    float* __restrict__ mem,            // (M x N) persistent membrane state
    float* __restrict__ spk_out,        // (M x N) spike output
    float* __restrict__ row0_trace,     // (N) spikes of batch row 0, or nullptr
    float* __restrict__ elem00_trace,   // mem[0,0], or nullptr
    int M, int K, int N)
{
    __shared__ __align__(16) float panel[2][64 * PROW];

    const int tid  = threadIdx.x;
    const int lane = tid & 31;
    const int wid  = tid >> 5;
    const int wave = blockIdx.x * 8 + wid;
    const int tilesM = M >> 4;
    const int tilesN = N >> 6;
    if (wave >= tilesM * tilesN) return;   // grids divide exactly; never splits a block

    const int m0   = (wave % tilesM) << 4;
    const int n0   = (wave / tilesM) << 6; // block-uniform (8 | 16 = tilesM)
    const int half = lane >> 4;            // K-half select (ISA f32 A/B layout)
    const int r    = lane & 15;

    // A fragment source: row m0+r, float2 at k + 2*half (A VGPR0/1 = K pair)
    const float* __restrict__ arow = A + (size_t)(m0 + r) * K + 2 * half;

    const unsigned lb0 = (unsigned)(size_t)&panel[0][0];
    const unsigned lb1 = (unsigned)(size_t)&panel[1][0];

    // Prime first panel.
    panel_load_async(W, K, n0, 0, lb0, tid);
    asm volatile("s_wait_asynccnt 0x0" ::: "memory");
    __syncthreads();

    v8f acc0 = {}, acc1 = {}, acc2 = {}, acc3 = {};

    int cur = 0;
    for (int k0 = 0; k0 < K; k0 += KC) {
        // Kick off next panel into the other buffer while we compute this one.
        if (k0 + KC < K)
            panel_load_async(W, K, n0, k0 + KC, cur ? lb0 : lb1, tid);

        const float* lp = &panel[cur][0];
        // B fragment rows: panel row (n - n0) = 16j + r, float2 at col kk + 2*half
        const float* l0 = lp + (r +  0) * PROW + 2 * half;
        const float* l1 = lp + (r + 16) * PROW + 2 * half;
        const float* l2 = lp + (r + 32) * PROW + 2 * half;
        const float* l3 = lp + (r + 48) * PROW + 2 * half;
        const float* aK = arow + k0;

#pragma unroll
        for (int kk = 0; kk < KC; kk += 4) {
            v2f a  = *(const v2f*)(aK + kk);
            v2f b0 = *(const v2f*)(l0 + kk);
            v2f b1 = *(const v2f*)(l1 + kk);
            v2f b2 = *(const v2f*)(l2 + kk);
            v2f b3 = *(const v2f*)(l3 + kk);
            acc0 = __builtin_amdgcn_wmma_f32_16x16x4_f32(false, a, false, b0, (short)0, acc0, false, false);
            acc1 = __builtin_amdgcn_wmma_f32_16x16x4_f32(false, a, false, b1, (short)0, acc1, false, false);
            acc2 = __builtin_amdgcn_wmma_f32_16x16x4_f32(false, a, false, b2, (short)0, acc2, false, false);
            acc3 = __builtin_amdgcn_wmma_f32_16x16x4_f32(false, a, false, b3, (short)0, acc3, false, false);
        }

        // Next panel must have landed, and all waves must be done reading `cur`.
        asm volatile("s_wait_asynccnt 0x0" ::: "memory");
        __syncthreads();
        cur ^= 1;
    }

    const float beta = *beta_p;
    v8f accs[4] = {acc0, acc1, acc2, acc3};

#pragma unroll
    for (int j = 0; j < 4; ++j) {
        const int   n  = n0 + 16 * j + r;
        const float bv = bias[n];
#pragma unroll
        for (int v = 0; v < 8; ++v) {
            const int m = m0 + v + 8 * half;   // C/D layout: VGPR v -> row v / v+8
            const float cur1  = accs[j][v] + bv;
            const size_t idx  = (size_t)m * N + n;
            const float mold  = mem[idx];
            const float reset = (mold - THRESH > 0.0f) ? THRESH : 0.0f; // reset uses prev mem
            const float mnew  = beta * mold + cur1 - reset;
            const float s     = (mnew - THRESH > 0.0f) ? 1.0f : 0.0f;
            mem[idx]     = mnew;
            spk_out[idx] = s;
            if (row0_trace && m == 0) {
                row0_trace[n] = s;
                if (elem00_trace && n == 0) *elem00_trace = mnew;
            }
        }
    }
}

extern "C" void kernel_launch(void* const* d_in, const int* in_sizes, int n_in,
                              void* d_out, int out_size, void* d_ws, size_t ws_size,
                              hipStream_t stream) {
    const float* x   = (const float*)d_in[0];
    const float* W1  = (const float*)d_in[1];
    const float* b1  = (const float*)d_in[2];
    const float* W2  = (const float*)d_in[3];
    const float* b2  = (const float*)d_in[4];
    // d_in[5] = num_steps (device int, unused on host)
    const float* beta1p = (const float*)d_in[6];
    const float* beta2p = (const float*)d_in[7];

    // Derive dims: H from b1, NI from W1, O from b2;
    // out_size = T*H + T + (T*B)*O with T*B = |x|/NI.
    const long H  = in_sizes[2];
    const long NI = (long)in_sizes[1] / H;
    const long O  = in_sizes[4];
    const long TB = (long)in_sizes[0] / NI;            // T * B
    const long T  = ((long)out_size - TB * O) / (H + 1);
    const long B  = TB / T;

    float* ws   = (float*)d_ws;
    float* mem1 = ws;                       // B*H
    float* mem2 = ws + B * H;               // B*O
    float* spk1 = ws + B * H + B * O;       // B*H

    float* out        = (float*)d_out;
    float* spk1_trace = out;                // T*H
    float* mem1_trace = out + T * H;        // T
    float* spk2_rec   = mem1_trace + T;     // T*B*O

    // Zero membrane state (mem1 and mem2 contiguous in ws).
    const long stateN = B * H + B * O;
    snn_zero_state<<<(int)((stateN + 255) / 256), 256, 0, stream>>>(ws, stateN);

    const int waves1  = (int)((B / 16) * (H / 64));
    const int blocks1 = (waves1 + 7) / 8;               // 8 waves / 256-thread block
    const int waves2  = (int)((B / 16) * (O / 64));
    const int blocks2 = (waves2 + 7) / 8;

    for (long t = 0; t < T; ++t) {
        // Layer 1: cur1 = x_t @ W1^T + b1 ; LIF ; spk1 -> ws, traces -> d_out
        snn_lif_gemm_step<<<blocks1, 256, 0, stream>>>(
            x + t * B * NI, W1, b1, beta1p,
            mem1, spk1,
            spk1_trace + t * H, mem1_trace + t,
            (int)B, (int)NI, (int)H);

        // Layer 2: cur2 = spk1 @ W2^T + b2 ; LIF ; spk2 -> d_out
        snn_lif_gemm_step<<<blocks2, 256, 0, stream>>>(
            spk1, W2, b2, beta2p,
            mem2, spk2_rec + t * B * O,
            nullptr, nullptr,
            (int)B, (int)H, (int)O);
    }
}